// glocapBlockAttention_42631845380376
// MI455X (gfx1250) — compile-verified
//
#include <hip/hip_runtime.h>
#include <hip/hip_bf16.h>

typedef __attribute__((ext_vector_type(2))) float v2f;
typedef __attribute__((ext_vector_type(8))) float v8f;

#define N_   32
#define C_   32
#define K_   10      // capsule classes
#define KP   16      // K padded to WMMA tile
#define S_   256     // W*H
#define E_   16      // P*P
#define LRS  257     // lrT LDS stride (pad to break bank conflicts)
#define BS   17      // b/c tile LDS stride
#define NROUT 3

__launch_bounds__(256, 1)
__global__ void capsule_route_kernel(const float* __restrict__ lg,
                                     const float* __restrict__ g,
                                     const float* __restrict__ wgt,
                                     float* __restrict__ out)
{
    __shared__ float lrT[E_ * LRS];        // [ij][s]   transposed lr tile for one (n,c)
    __shared__ float wt_s[C_ * K_ * E_];   // [c][k][j*4+d]  all pose weights (20 KB)
    __shared__ float bC[S_ * BS];          // [s][k]    b logits, then softmax c (cols 10..15 = 0)
    __shared__ float U_s[KP * E_];         // [k][i*4+j]  per-c projection of q through wt
    __shared__ float q_s[KP * E_];         // [k][e]    current routing vector (g or gc), padded
    __shared__ float sacc[KP * E_];        // [k][e]    s accumulator over c
    __shared__ float Mfull[E_ * KP];       // [ij][k]   lr^T @ c, reduced across waves
    __shared__ float coef_s[KP];

    const int tid  = threadIdx.x;
    const int lane = tid & 31;
    const int wave = tid >> 5;
    const int half = (lane >> 4) & 1;      // which 16-lane half of the wave
    const int l16  = lane & 15;
    const int n    = blockIdx.x;

    // Stage all weights once: 32*10*16 floats
    for (int idx = tid; idx < C_ * K_ * E_; idx += 256) wt_s[idx] = wgt[idx];
    // q <- g[n], zero-padded to 16 rows
    if (tid < KP * E_) q_s[tid] = (tid < K_ * E_) ? g[n * K_ * E_ + tid] : 0.0f;
    __syncthreads();

    for (int t = 0; t < NROUT; ++t) {
        if (tid < KP * E_) sacc[tid] = 0.0f;
        __syncthreads();

        for (int c = 0; c < C_; ++c) {
            const float* lbase = lg + ((size_t)(n * C_ + c) * E_) * S_;

            // ---- stage lr tile (transposed): lrT[ij][s] = l[n,c,ij,s] ----
            for (int idx = tid; idx < E_ * S_; idx += 256) {
                const int ij = idx >> 8, s = idx & 255;
                lrT[ij * LRS + s] = lbase[ij * S_ + s];
            }
            if (c + 1 < C_)  // hint next tile toward L2/L0
                __builtin_prefetch(lbase + (size_t)E_ * S_ + tid * 16, 0, 1);

            // ---- U[k][i*4+j] = sum_d wt[c,k,j,d] * q[k][i*4+d]  (pad rows zeroed) ----
            if (tid < KP * E_) {
                if (tid < K_ * E_) {
                    const int k = tid >> 4, e = tid & 15;
                    const int i = e >> 2, j = e & 3;
                    float acc = 0.0f;
                    #pragma unroll
                    for (int d = 0; d < 4; ++d)
                        acc += wt_s[(c * K_ + k) * E_ + j * 4 + d] * q_s[k * E_ + i * 4 + d];
                    U_s[tid] = acc;
                } else {
                    U_s[tid] = 0.0f;
                }
            }
            __syncthreads();

            // ---- b-phase: b[s,k] = lr(256x16) @ U^T(16x16), 16 row-chunks over 8 waves ----
            #pragma unroll
            for (int mi = 0; mi < 2; ++mi) {
                const int m = wave + mi * 8;       // row chunk: s = m*16 .. m*16+15
                v8f acc = {};
                #pragma unroll
                for (int kk = 0; kk < 4; ++kk) {   // reduction over ij in steps of 4
                    v2f a, b;
                    a[0] = lrT[(kk * 4 + 0 + 2 * half) * LRS + m * 16 + l16];
                    a[1] = lrT[(kk * 4 + 1 + 2 * half) * LRS + m * 16 + l16];
                    b[0] = U_s[l16 * E_ + kk * 4 + 0 + 2 * half];
                    b[1] = U_s[l16 * E_ + kk * 4 + 1 + 2 * half];
                    acc = __builtin_amdgcn_wmma_f32_16x16x4_f32(
                            false, a, false, b, (short)0, acc, false, false);
                }
                #pragma unroll
                for (int v = 0; v < 8; ++v)
                    bC[(m * 16 + v + 8 * half) * BS + l16] = acc[v];
            }
            __syncthreads();

            // ---- softmax over k (rows of bC), and zero Mfull for this c ----
            {
                float* row = &bC[tid * BS];
                float mx = row[0];
                #pragma unroll
                for (int k = 1; k < K_; ++k) mx = fmaxf(mx, row[k]);
                float e[K_], sum = 0.0f;
                #pragma unroll
                for (int k = 0; k < K_; ++k) { e[k] = __expf(row[k] - mx); sum += e[k]; }
                const float inv = 1.0f / sum;
                #pragma unroll
                for (int k = 0; k < K_; ++k) row[k] = e[k] * inv;
                #pragma unroll
                for (int k = K_; k < KP; ++k) row[k] = 0.0f;
                Mfull[tid] = 0.0f;
            }
            __syncthreads();

            // ---- M-phase: M[ij,k] = lr^T(16x256) @ c(256x16), K split 8 chunks/wave ----
            {
                v8f macc = {};
                for (int sc = wave * 8; sc < wave * 8 + 8; ++sc) {
                    const int s0 = sc * 4;
                    v2f a, b;
                    a[0] = lrT[l16 * LRS + s0 + 0 + 2 * half];
                    a[1] = lrT[l16 * LRS + s0 + 1 + 2 * half];
                    b[0] = bC[(s0 + 0 + 2 * half) * BS + l16];
                    b[1] = bC[(s0 + 1 + 2 * half) * BS + l16];
                    macc = __builtin_amdgcn_wmma_f32_16x16x4_f32(
                             false, a, false, b, (short)0, macc, false, false);
                }
                #pragma unroll
                for (int v = 0; v < 8; ++v)
                    atomicAdd(&Mfull[(v + 8 * half) * KP + l16], macc[v]);  // ds_add_f32
            }
            __syncthreads();

            // ---- s[k][i*4+d] += sum_j M[(i*4+j),k] * wt[c,k,j,d] ----
            if (tid < K_ * E_) {
                const int k = tid >> 4, e = tid & 15;
                const int i = e >> 2, d = e & 3;
                float acc = 0.0f;
                #pragma unroll
                for (int j = 0; j < 4; ++j)
                    acc += Mfull[(i * 4 + j) * KP + k] * wt_s[(c * K_ + k) * E_ + j * 4 + d];
                sacc[tid] += acc;
            }
            __syncthreads();
        } // c

        // ---- squash: gc = s * sqrt(|s|^2) / (1 + |s|^2) ----
        if (tid < K_) {
            float sn = 0.0f;
            #pragma unroll
            for (int e = 0; e < E_; ++e) { const float v = sacc[tid * E_ + e]; sn += v * v; }
            coef_s[tid] = sqrtf(sn) / (1.0f + sn);
        }
        __syncthreads();
        if (tid < KP * E_)
            q_s[tid] = (tid < K_ * E_) ? sacc[tid] * coef_s[tid >> 4] : 0.0f;
        __syncthreads();
    } // rounds

    // ---- outputs: a (N*K) then gc (N*K*16), concatenated flat ----
    if (tid < K_ * E_) {
        const int k = tid >> 4, e = tid & 15;
        out[N_ * K_ + (n * K_ + k) * E_ + e] = q_s[tid];
    }
    if (tid < K_) {
        float gn = 0.0f;
        #pragma unroll
        for (int e = 0; e < E_; ++e) { const float v = q_s[tid * E_ + e]; gn += v * v; }
        const float r = sqrtf(gn);
        out[n * K_ + tid] = 1.0f / (1.0f + __expf(-r));
    }
}

extern "C" void kernel_launch(void* const* d_in, const int* in_sizes, int n_in,
                              void* d_out, int out_size, void* d_ws, size_t ws_size,
                              hipStream_t stream) {
    const float* l = (const float*)d_in[0];   // (32,32,16,16,16) fp32
    const float* g = (const float*)d_in[1];   // (32,10,16) fp32
    const float* w = (const float*)d_in[2];   // (1,32,1,10,4,4) fp32
    (void)in_sizes; (void)n_in; (void)out_size; (void)d_ws; (void)ws_size;
    // d_in[3] = n_rout (==3, baked in as NROUT)
    capsule_route_kernel<<<N_, 256, 0, stream>>>(l, g, w, (float*)d_out);
}